// STGCN_45174466019757
// MI455X (gfx1250) — compile-verified
//
#include <hip/hip_runtime.h>
#include <hip/hip_bf16.h>
#include <math.h>

typedef float v2f __attribute__((ext_vector_type(2)));
typedef float v8f __attribute__((ext_vector_type(8)));

#define GM_ACC  1
#define GM_BIAS 2
#define GM_RELU 4

// D = A(16x4,f32) * B(4x16,f32) + C ; full f32 WMMA on CDNA5
__device__ inline v8f wmma4(v2f a, v2f b, v8f c) {
  return __builtin_amdgcn_wmma_f32_16x16x4_f32(false, a, false, b, (short)0, c,
                                               false, false);
}

// CDNA5 async global->LDS copy (ASYNCcnt path, bypasses VGPRs).
// dsaddr = low 32 bits of the generic LDS pointer (wave-relative LDS offset).
__device__ inline void async_ld_b128(float* ldst, const float* gsrc) {
  unsigned lofs = (unsigned)(unsigned long long)ldst;
  asm volatile("global_load_async_to_lds_b128 %0, %1, off"
               :: "v"(lofs), "v"(gsrc) : "memory");
}
__device__ inline void async_wait0() {
  asm volatile("s_wait_asynccnt 0x0" ::: "memory");
}

// ---------------- utility kernels ----------------
__global__ void k_zero(float* p, long long n) {
  long long i = (long long)blockIdx.x * blockDim.x + threadIdx.x;
  long long stride = (long long)gridDim.x * blockDim.x;
  for (; i < n; i += stride) p[i] = 0.0f;
}

__global__ void k_deg(const int* __restrict__ src, float* __restrict__ deg, int E) {
  int e = blockIdx.x * blockDim.x + threadIdx.x;
  if (e < E) atomicAdd(&deg[src[e]], 1.0f);
}

__global__ void k_ew(const int* __restrict__ src, const int* __restrict__ dst,
                     const float* __restrict__ deg, float* __restrict__ wE, int E) {
  int e = blockIdx.x * blockDim.x + threadIdx.x;
  if (e >= E) return;
  float ds = deg[src[e]], dd = deg[dst[e]];
  float is = ds > 0.0f ? rsqrtf(fmaxf(ds, 1.0f)) : 0.0f;
  float id = dd > 0.0f ? rsqrtf(fmaxf(dd, 1.0f)) : 0.0f;
  wE[e] = -(is * id);
}

// Tx2 = 2*acc - tx0  (in place on acc)
__global__ void k_tx2(float* __restrict__ acc, const float* __restrict__ tx0,
                      long long n) {
  long long i = (long long)blockIdx.x * blockDim.x + threadIdx.x;
  long long stride = (long long)gridDim.x * blockDim.x;
  for (; i < n; i += stride) acc[i] = 2.0f * acc[i] - tx0[i];
}

// sparse L_hat @ v : scatter-add over edges with atomicAdd
__global__ void k_mv(const float* __restrict__ V, const int* __restrict__ src,
                     const int* __restrict__ dst, const float* __restrict__ wE,
                     float* __restrict__ Out, int BT, int N, int C, int E) {
  long long idx = (long long)blockIdx.x * blockDim.x + threadIdx.x;
  long long total = (long long)BT * E * C;
  if (idx >= total) return;
  int c = (int)(idx % C);
  long long t = idx / C;
  int e = (int)(t % E);
  int bt = (int)(t / E);
  float w = wE[e];
  float val = w * V[((long long)bt * N + src[e]) * C + c];
  atomicAdd(&Out[((long long)bt * N + dst[e]) * C + c], val);
}

// ---------------- generic 16-row WMMA GEMM ----------------
// Out[r0:r0+16, co0:co0+16] (op)= A[r0:r0+16, 0:K] @ W[0:K, co0:co0+16]
// blockDim = 32 (one wave32); grid.x = R/16, grid.y = Co/16
__global__ void k_gemm16(const float* __restrict__ A, const float* __restrict__ W,
                         const float* __restrict__ bias, float* __restrict__ Out,
                         int K, int Co, int mode) {
  extern __shared__ __align__(16) float lds[];  // 16*K floats
  long long r0 = (long long)blockIdx.x * 16;
  int co0 = blockIdx.y * 16;
  int lane = threadIdx.x;
  // async-stage the 16xK A tile straight into LDS (B128 quads)
  int QK = K >> 2;  // quads per row
  for (int q = lane; q < 16 * QK; q += 32) {
    int m = q / QK, kk = (q - m * QK) << 2;
    async_ld_b128(lds + m * K + kk, A + (r0 + m) * K + kk);
  }
  async_wait0();
  __syncthreads();
  int m = lane & 15;              // A row / B col within tile
  int kb = (lane >> 4) << 1;      // K sub-offset: 0 or 2
  v8f acc = {};
  for (int k = 0; k < K; k += 4) {
    v2f a, b;
    a.x = lds[m * K + k + kb];
    a.y = lds[m * K + k + kb + 1];
    b.x = W[(k + kb) * Co + co0 + m];
    b.y = W[(k + kb + 1) * Co + co0 + m];
    acc = wmma4(a, b, acc);
  }
  int colg = co0 + m;
  float bv = (mode & GM_BIAS) ? bias[colg] : 0.0f;
  int rbase = (lane >> 4) * 8;    // C/D layout: VGPR v -> M=v (+8 for hi half)
  for (int v = 0; v < 8; ++v) {
    long long o = (r0 + rbase + v) * Co + colg;
    float x = acc[v] + bv;
    if (mode & GM_ACC) x += Out[o];
    if (mode & GM_RELU) x = fmaxf(x, 0.0f);
    Out[o] = x;
  }
}

// ---------------- gated temporal conv (KS=3), fused WMMA ----------------
// X (B,Tin,N,Ci) -> Out (B,Tin-2,N,Co); Out = relu(P*sigmoid(Q)+R)
// blockDim = 32; grid.x = B*(Tin-2)*(N/16), grid.y = Co/16
__global__ void k_tconv(const float* __restrict__ X,
                        const float* __restrict__ W1, const float* __restrict__ B1,
                        const float* __restrict__ W2, const float* __restrict__ B2,
                        const float* __restrict__ W3, const float* __restrict__ B3,
                        float* __restrict__ Out,
                        int Bb, int Tin, int N, int Ci, int Co) {
  extern __shared__ __align__(16) float lds[];  // 16 * (3*Ci) floats
  int K = 3 * Ci;
  int Tout = Tin - 2;
  int ntiles = N >> 4;
  int tile = blockIdx.x;
  int n0 = (tile % ntiles) << 4;
  int t = (tile / ntiles) % Tout;
  int bidx = tile / (ntiles * Tout);
  int co0 = blockIdx.y * 16;
  int lane = threadIdx.x;
  // im2col gather via async B128: A[m][kt*Ci+c..c+3] = X[b, t+kt, n0+m, c..c+3]
  int QK = K >> 2;
  for (int q = lane; q < 16 * QK; q += 32) {
    int m = q / QK, kk = (q - m * QK) << 2;
    int kt = kk / Ci, c = kk - kt * Ci;
    async_ld_b128(lds + m * K + kk,
                  X + (((long long)bidx * Tin + t + kt) * N + n0 + m) * Ci + c);
  }
  async_wait0();
  __syncthreads();
  int m = lane & 15;
  int kb = (lane >> 4) << 1;
  v8f aP = {}, aQ = {}, aR = {};
  for (int k = 0; k < K; k += 4) {
    v2f a;
    a.x = lds[m * K + k + kb];
    a.y = lds[m * K + k + kb + 1];
    int w0 = (k + kb) * Co + co0 + m;
    int w1 = w0 + Co;
    v2f b;
    b.x = W1[w0]; b.y = W1[w1];
    aP = wmma4(a, b, aP);
    b.x = W2[w0]; b.y = W2[w1];
    aQ = wmma4(a, b, aQ);
    b.x = W3[w0]; b.y = W3[w1];
    aR = wmma4(a, b, aR);
  }
  int colg = co0 + m;
  float bp = B1[colg], bq = B2[colg], br = B3[colg];
  int rbase = (lane >> 4) * 8;
  for (int v = 0; v < 8; ++v) {
    float p = aP[v] + bp, q = aQ[v] + bq, r = aR[v] + br;
    float g = 1.0f / (1.0f + __expf(-q));
    float y = fmaxf(p * g + r, 0.0f);
    Out[(((long long)bidx * Tout + t) * N + n0 + rbase + v) * Co + colg] = y;
  }
}

// ---------------- per-node BatchNorm (training stats) + relu ----------------
// H (B,T,N,C); stats per node over (B,T,C); one block per node
__global__ void k_bn(const float* __restrict__ H, const float* __restrict__ g,
                     const float* __restrict__ be, float* __restrict__ Out,
                     int Bb, int T, int N, int C) {
  __shared__ float s1[256], s2[256];
  int n = blockIdx.x;
  int tid = threadIdx.x;
  int cnt = Bb * T * C;
  float sum = 0.0f, sq = 0.0f;
  for (int i = tid; i < cnt; i += 256) {
    int c = i % C;
    int bt = i / C;
    float x = H[((long long)bt * N + n) * C + c];
    sum += x;
    sq += x * x;
  }
  s1[tid] = sum;
  s2[tid] = sq;
  __syncthreads();
  for (int s = 128; s > 0; s >>= 1) {
    if (tid < s) { s1[tid] += s1[tid + s]; s2[tid] += s2[tid + s]; }
    __syncthreads();
  }
  float mean = s1[0] / (float)cnt;
  float var = s2[0] / (float)cnt - mean * mean;
  float scale = rsqrtf(var + 1e-5f) * g[n];
  float shift = be[n];
  for (int i = tid; i < cnt; i += 256) {
    int c = i % C;
    int bt = i / C;
    long long o = ((long long)bt * N + n) * C + c;
    Out[o] = fmaxf((H[o] - mean) * scale + shift, 0.0f);
  }
}

// ---------------- head FC: Out(16 x N) = H(16 x K) @ Wfc^T + bfc -----------
// K = N*C = 32768; the 268MB Wfc stream is the bandwidth-critical path.
// block = 512 threads (16 waves); each wave covers K/16; LDS reduce.
#define FC_WAVES 16
__global__ void k_fc(const float* __restrict__ H, const float* __restrict__ Wfc,
                     const float* __restrict__ bfc, float* __restrict__ Out,
                     int Kdim, int N) {
  __shared__ float red[FC_WAVES * 32 * 8];
  int n0 = blockIdx.x * 16;
  int tid = threadIdx.x;
  int w = tid >> 5, lane = tid & 31;
  int m = lane & 15;
  int kb = (lane >> 4) << 1;
  int kspan = Kdim / FC_WAVES;
  int k0 = w * kspan, k1 = k0 + kspan;
  const float* arow = H + (long long)m * Kdim;              // batch row m
  const float* wrow = Wfc + (long long)(n0 + m) * Kdim;     // Wfc row = B col
  v8f acc = {};
  for (int k = k0; k < k1; k += 4) {
    if ((k & 255) == 0) __builtin_prefetch(wrow + k + 256, 0, 1);
    v2f a = *(const v2f*)(arow + k + kb);
    v2f b = *(const v2f*)(wrow + k + kb);
    acc = wmma4(a, b, acc);
  }
  for (int i = 0; i < 8; ++i) red[(w * 32 + lane) * 8 + i] = acc[i];
  __syncthreads();
  if (tid < 32) {
    v8f tot = {};
    for (int ww = 0; ww < FC_WAVES; ++ww)
      for (int i = 0; i < 8; ++i) tot[i] += red[(ww * 32 + lane) * 8 + i];
    float bv = bfc[n0 + m];
    int rbase = (lane >> 4) * 8;
    for (int v = 0; v < 8; ++v)
      Out[(long long)(rbase + v) * N + n0 + m] = tot[v] + bv;
  }
}

// ---------------- host-side ST-Conv block ----------------
static void run_stblock(void* const* d_in, int pi, const float* Xin, int Tin,
                        const float* wE, const int* src, const int* dst,
                        float* bufA, float* bufB, float* bufC, float* bufD,
                        float* Out, hipStream_t stream) {
  const int B = 16, N = 2048, E = 32768;
  const float* k1 = (const float*)d_in[pi + 0];
  const float* b1 = (const float*)d_in[pi + 1];
  const float* k2 = (const float*)d_in[pi + 2];
  const float* b2 = (const float*)d_in[pi + 3];
  const float* k3 = (const float*)d_in[pi + 4];
  const float* b3 = (const float*)d_in[pi + 5];
  const float* cW = (const float*)d_in[pi + 6];
  const float* cB = (const float*)d_in[pi + 7];
  const float* u1 = (const float*)d_in[pi + 8];
  const float* y1 = (const float*)d_in[pi + 9];
  const float* u2 = (const float*)d_in[pi + 10];
  const float* y2 = (const float*)d_in[pi + 11];
  const float* u3 = (const float*)d_in[pi + 12];
  const float* y3 = (const float*)d_in[pi + 13];
  const float* bng = (const float*)d_in[pi + 14];
  const float* bnb = (const float*)d_in[pi + 15];

  int Tout1 = Tin - 2, Tout2 = Tin - 4;
  long long rows1 = (long long)B * Tout1 * N;

  // gated temporal conv 1: Ci=16 -> Co=32
  k_tconv<<<dim3(B * Tout1 * (N / 16), 2), 32, 16 * 48 * 4, stream>>>(
      Xin, k1, b1, k2, b2, k3, b3, bufA, B, Tin, N, 16, 32);

  // ChebConv K=3: out = Tx0@W0 + Tx1@W1 + Tx2@W2 + b, then relu
  k_gemm16<<<dim3((unsigned)(rows1 / 16), 2), 32, 16 * 32 * 4, stream>>>(
      bufA, cW, nullptr, bufC, 32, 32, 0);
  k_zero<<<4096, 256, 0, stream>>>(bufB, rows1 * 32);
  {
    long long tot = (long long)B * Tout1 * E * 32;
    k_mv<<<(unsigned)((tot + 255) / 256), 256, 0, stream>>>(
        bufA, src, dst, wE, bufB, B * Tout1, N, 32, E);
  }
  k_gemm16<<<dim3((unsigned)(rows1 / 16), 2), 32, 16 * 32 * 4, stream>>>(
      bufB, cW + 1024, nullptr, bufC, 32, 32, GM_ACC);
  k_zero<<<4096, 256, 0, stream>>>(bufD, rows1 * 32);
  {
    long long tot = (long long)B * Tout1 * E * 32;
    k_mv<<<(unsigned)((tot + 255) / 256), 256, 0, stream>>>(
        bufB, src, dst, wE, bufD, B * Tout1, N, 32, E);
  }
  k_tx2<<<4096, 256, 0, stream>>>(bufD, bufA, rows1 * 32);
  k_gemm16<<<dim3((unsigned)(rows1 / 16), 2), 32, 16 * 32 * 4, stream>>>(
      bufD, cW + 2048, cB, bufC, 32, 32, GM_ACC | GM_BIAS | GM_RELU);

  // gated temporal conv 2: Ci=32 -> Co=16
  k_tconv<<<dim3(B * Tout2 * (N / 16), 1), 32, 16 * 96 * 4, stream>>>(
      bufC, u1, y1, u2, y2, u3, y3, bufA, B, Tout1, N, 32, 16);

  // per-node BatchNorm + outer relu
  k_bn<<<N, 256, 0, stream>>>(bufA, bng, bnb, Out, B, Tout2, N, 16);
}

extern "C" void kernel_launch(void* const* d_in, const int* in_sizes, int n_in,
                              void* d_out, int out_size, void* d_ws, size_t ws_size,
                              hipStream_t stream) {
  const int B = 16, T0 = 9, N = 2048, C = 16, E = 32768;
  (void)in_sizes; (void)n_in; (void)out_size; (void)ws_size;

  const float* x   = (const float*)d_in[0];
  const float* Wf  = (const float*)d_in[33];
  const float* bf  = (const float*)d_in[34];
  const float* Wfc = (const float*)d_in[35];
  const float* bfc = (const float*)d_in[36];
  const int* ei    = (const int*)d_in[37];
  const int* src = ei;
  const int* dst = ei + E;

  float* w = (float*)d_ws;
  float* deg = w;  w += N;
  float* wE  = w;  w += E;
  const long long CAP = (long long)B * 7 * N * 32;  // largest intermediate
  float* buf1 = w; w += CAP;
  float* buf2 = w; w += CAP;
  float* buf3 = w; w += CAP;
  float* buf4 = w; w += CAP;

  // Cheb edge weights: w_e = -(deg_src^-1/2 * deg_dst^-1/2)
  k_zero<<<64, 256, 0, stream>>>(deg, N);
  k_deg<<<(E + 255) / 256, 256, 0, stream>>>(src, deg, E);
  k_ew<<<(E + 255) / 256, 256, 0, stream>>>(src, dst, deg, wE, E);

  // ST block 1: x (B,9,N,16) -> buf2 (B,5,N,16)
  run_stblock(d_in, 1, x, T0, wE, src, dst, buf1, buf2, buf3, buf4, buf2, stream);
  // ST block 2: buf2 (B,5,N,16) -> buf3 (B,1,N,16)
  run_stblock(d_in, 17, buf2, 5, wE, src, dst, buf1, buf4, buf3, buf2, buf3, stream);

  // head: 1x1 channel projection (B*N x 16) @ Wf(16x16) + bf -> buf1
  k_gemm16<<<dim3(B * N / 16, 1), 32, 16 * 16 * 4, stream>>>(
      buf3, Wf, bf, buf1, 16, 16, GM_BIAS);

  // head FC: (16 x 32768) @ Wfc^T + bfc -> d_out (16 x 2048)
  k_fc<<<N / 16, 512, 0, stream>>>(buf1, Wfc, bfc, (float*)d_out, N * C, N);
}